// Retrace_31894427140439
// MI455X (gfx1250) — compile-verified
//
#include <hip/hip_runtime.h>
#include <stdint.h>

#ifndef __has_builtin
#define __has_builtin(x) 0
#endif

#define GAMMA 0.99f

constexpr int S_LEN = 1025;   // sequence length
constexpr int COLS  = 1024;   // S-1 loss columns
constexpr int NSCAN = 1023;   // S-2 scan elements

// ---------- affine map q -> a*q + b, composed right-to-left (suffix scan) ----
struct Aff { float a; float b; };

__device__ __forceinline__ Aff aff_compose(const Aff x, const Aff y) {
  // (x o y)(q) = x.a*(y.a*q + y.b) + x.b
  Aff r;
  r.a = x.a * y.a;
  r.b = fmaf(x.a, y.b, x.b);
  return r;
}

// Inclusive suffix scan across a wave32: lane l gets f_l o f_{l+1} o ... o f_31
__device__ __forceinline__ Aff wave_suffix_scan(Aff f) {
  const int lane = threadIdx.x & 31;
#pragma unroll
  for (int off = 1; off < 32; off <<= 1) {
    float ta = __shfl_down(f.a, (unsigned)off, 32);
    float tb = __shfl_down(f.b, (unsigned)off, 32);
    if (lane + off < 32) {
      Aff t; t.a = ta; t.b = tb;
      f = aff_compose(f, t);
    }
  }
  return f;
}

// ---------------- CDNA5 async global->LDS path (guarded) --------------------
#if __has_builtin(__builtin_amdgcn_global_load_async_to_lds_b32)
#define RT_ASYNC_LDS 1
typedef __attribute__((address_space(1))) int* as1ip;   // global int*
typedef __attribute__((address_space(3))) int* as3ip;   // LDS int*
__device__ __forceinline__ void async_copy_f32(const float* g, float* l) {
  __builtin_amdgcn_global_load_async_to_lds_b32((as1ip)(g), (as3ip)(l), 0, 0);
}
__device__ __forceinline__ void wait_async_zero() {
#if __has_builtin(__builtin_amdgcn_s_wait_asynccnt)
  __builtin_amdgcn_s_wait_asynccnt(0);
#else
  asm volatile("s_wait_asynccnt 0" ::: "memory");
#endif
}
#else
#define RT_ASYNC_LDS 0
#endif

// ---------------- kernel 1: one row per block ------------------------------
__global__ __launch_bounds__(1024) void retrace_rows(
    const float* __restrict__ Qm,   // Q
    const float* __restrict__ eQ,   // expected_target_Q
    const float* __restrict__ tQ,   // target_Q
    const float* __restrict__ rw,   // rewards
    const float* __restrict__ tp,   // target_policy_probs
    const float* __restrict__ bp,   // behaviour_policy_probs
    float* __restrict__ partial)    // per-row SSE
{
  const int row  = blockIdx.x;
  const int tid  = (int)threadIdx.x;       // 0..1023 == loss column
  const int lane = tid & 31;
  const int wv   = tid >> 5;               // 0..31
  const int64_t base = (int64_t)row * S_LEN;

  __shared__ float s_wa[32];
  __shared__ float s_wb[32];
  __shared__ float s_red[32];

  // scan element index, clamped so the last thread still issues valid loads
  const int t = (tid < NSCAN) ? tid : (NSCAN - 1);

  float q_v, r_v, e_v, t_v, p_v, b_v;

#if RT_ASYNC_LDS
  __shared__ float st_q[COLS];
  __shared__ float st_r[COLS];
  __shared__ float st_e[COLS];
  __shared__ float st_t[COLS];
  __shared__ float st_p[COLS];
  __shared__ float st_b[COLS];
  async_copy_f32(Qm + base + tid,   &st_q[tid]);
  async_copy_f32(rw + base + t + 1, &st_r[tid]);
  async_copy_f32(eQ + base + t + 2, &st_e[tid]);
  async_copy_f32(tQ + base + t + 2, &st_t[tid]);
  async_copy_f32(tp + base + t + 2, &st_p[tid]);
  async_copy_f32(bp + base + t + 2, &st_b[tid]);
  wait_async_zero();   // each lane reads back exactly what it DMA'd
  q_v = st_q[tid]; r_v = st_r[tid]; e_v = st_e[tid];
  t_v = st_t[tid]; p_v = st_p[tid]; b_v = st_b[tid];
#else
  q_v = Qm[base + tid];
  r_v = rw[base + t + 1];
  e_v = eQ[base + t + 2];
  t_v = tQ[base + t + 2];
  p_v = tp[base + t + 2];
  b_v = bp[base + t + 2];
#endif

  // importance weight and affine map for this scan element
  const float c = expf(fminf(p_v - b_v, 0.0f));
  Aff f;
  if (tid < NSCAN) {
    f.a = GAMMA * c;
    f.b = fmaf(GAMMA, e_v - c * t_v, r_v);   // r + g*(e - c*tq)
  } else {
    f.a = 1.0f; f.b = 0.0f;                  // identity padding (element 1023)
  }

  // ---- hierarchical suffix scan: intra-wave shuffles + inter-wave via LDS --
  Aff sw = wave_suffix_scan(f);              // covers [tid .. end of wave]
  if (lane == 0) { s_wa[wv] = sw.a; s_wb[wv] = sw.b; }
  __syncthreads();
  if (wv == 0) {
    Aff tw; tw.a = s_wa[lane]; tw.b = s_wb[lane];
    Aff inc = wave_suffix_scan(tw);          // inclusive over wave totals
    float ea = __shfl_down(inc.a, 1u, 32);   // exclusive = next lane's inclusive
    float eb = __shfl_down(inc.b, 1u, 32);
    if (lane == 31) { ea = 1.0f; eb = 0.0f; }
    s_wa[lane] = ea; s_wb[lane] = eb;
  }
  __syncthreads();

  Aff ew; ew.a = s_wa[wv]; ew.b = s_wb[wv];  // suffix of all later waves
  const Aff tot = aff_compose(sw, ew);       // suffix [tid .. 1022]

  const float q_last = tQ[base + (S_LEN - 1)];       // uniform -> scalar load
  float q_ret = fmaf(tot.a, q_last, tot.b);
  if (tid == COLS - 1) q_ret = q_last;               // last column is q_last

  // ---- squared error + deterministic block reduction ----------------------
  const float diff = q_v - q_ret;
  float d2 = diff * diff;
#pragma unroll
  for (int off = 16; off > 0; off >>= 1) d2 += __shfl_down(d2, (unsigned)off, 32);
  if (lane == 0) s_red[wv] = d2;
  __syncthreads();
  if (wv == 0) {
    float v = s_red[lane];
#pragma unroll
    for (int off = 16; off > 0; off >>= 1) v += __shfl_down(v, (unsigned)off, 32);
    if (lane == 0) partial[row] = v;
  }
}

// ---------------- kernel 2: deterministic final reduction ------------------
__global__ __launch_bounds__(256) void reduce_partials(
    const float* __restrict__ partial, int n, float* __restrict__ out)
{
  const int tid  = (int)threadIdx.x;
  const int lane = tid & 31;
  const int wv   = tid >> 5;
  __shared__ double s_red[8];

  double acc = 0.0;
  for (int i = tid; i < n; i += 256) acc += (double)partial[i];
#pragma unroll
  for (int off = 16; off > 0; off >>= 1) acc += __shfl_down(acc, (unsigned)off, 32);
  if (lane == 0) s_red[wv] = acc;
  __syncthreads();
  if (tid == 0) {
    double tot = 0.0;
#pragma unroll
    for (int w = 0; w < 8; ++w) tot += s_red[w];
    out[0] = (float)(tot / ((double)n * (double)COLS));
  }
}

// ---------------- host entry ------------------------------------------------
extern "C" void kernel_launch(void* const* d_in, const int* in_sizes, int n_in,
                              void* d_out, int out_size, void* d_ws, size_t ws_size,
                              hipStream_t stream) {
  const float* Qm = (const float*)d_in[0];
  const float* eQ = (const float*)d_in[1];
  const float* tQ = (const float*)d_in[2];
  const float* rw = (const float*)d_in[3];
  const float* tp = (const float*)d_in[4];
  const float* bp = (const float*)d_in[5];

  const int rows = in_sizes[0] / S_LEN;     // 8192
  float* partial = (float*)d_ws;            // rows * 4 bytes of scratch

  retrace_rows<<<rows, 1024, 0, stream>>>(Qm, eQ, tQ, rw, tp, bp, partial);
  reduce_partials<<<1, 256, 0, stream>>>(partial, rows, (float*)d_out);
}